// PEPEmbedding_20779051778717
// MI455X (gfx1250) — compile-verified
//
#include <hip/hip_runtime.h>
#include <stdint.h>

// PEP embedding-bag: out[b,:] = sum_j soft_threshold(V[x[b,j],:])
// Gather-bandwidth bound (~55 MB @ 23.3 TB/s). Uses CDNA5 TDM gather-mode
// (TENSOR_LOAD_TO_LDS with gather descriptors, TENSORcnt) to DMA 8 rows per
// instruction into LDS; 4-deep LDS buffering keeps 3 descriptors in flight
// per wave (4096 waves x 3 x 2KB = 24MB outstanding ~ BW*latency product)
// while the wave applies the soft-threshold and accumulates.

#define IDX_NUM 1000000
#define DIM 64
#define BATCH 4096
#define BAG 50
#define ROWS_PER_DESC 8
#define NDESC 7                 // 6 full descriptors of 8 rows + 1 of 2 rows
#define LAST_ROWS (BAG - ROWS_PER_DESC * (NDESC - 1))   // = 2
#define NBUF 4                  // LDS buffers per wave (3 TDM ops in flight)
#define WAVES_PER_BLOCK 8

typedef __attribute__((ext_vector_type(4))) unsigned int v4u;
typedef __attribute__((ext_vector_type(8))) int          v8i;
typedef __attribute__((ext_vector_type(4))) int          v4i;

__device__ __forceinline__ void tdm_gather8(const float* vbase,
                                            uint32_t lds_byte_addr,
                                            const int* ids, int nvalid) {
  const uint64_t ga = (uint64_t)(uintptr_t)vbase;

  // D# group 0: count=1, is_restore=0, gather_index_size=1 (32-bit),
  // gather_mode=1, lds_addr, global_addr[56:0], type=2 ("image").
  v4u g0;
  g0.x = 1u | (1u << 30) | (1u << 31);
  g0.y = lds_byte_addr;
  g0.z = (uint32_t)(ga & 0xFFFFFFFFu);
  g0.w = (uint32_t)((ga >> 32) & 0x01FFFFFFu) | (2u << 30);

  // D# group 1: workgroup_mask=0, data_size=2 (4 bytes), no barrier/iterate/pad,
  // tensor_dim0=64, tensor_dim1=IDX_NUM, tile_dim0=64,
  // tile_dim1=#valid indices, tensor_dim0_stride=64.
  v8i g1;
  g1[0] = (int)(2u << 16);                                     // data_size=4B
  g1[1] = (int)((uint32_t)(DIM & 0xFFFF) << 16);               // tensor_dim0 lo16 @ [31:16]
  g1[2] = (int)(((uint32_t)IDX_NUM & 0xFFFFu) << 16);          // dim0 hi16=0 | tensor_dim1 lo16
  g1[3] = (int)(((uint32_t)IDX_NUM >> 16) & 0xFFFFu)           // tensor_dim1 hi16
        | (int)((uint32_t)DIM << 16);                          // tile_dim0=64
  g1[4] = nvalid & 0xFFFF;                                     // tile_dim1 = #valid rows
  g1[5] = DIM;                                                 // dim0_stride lo32 = 64
  g1[6] = 0;                                                   // dim0_stride hi16 | dim1_stride lo16 (ignored)
  g1[7] = 0;

  // D# groups 2+3: eight 32-bit row indices.
  v4i g2, g3;
  g2[0] = ids[0]; g2[1] = ids[1]; g2[2] = ids[2]; g2[3] = ids[3];
  g3[0] = ids[4]; g3[1] = ids[5]; g3[2] = ids[6]; g3[3] = ids[7];

#if __clang_major__ >= 23
  v8i gz = {};
  __builtin_amdgcn_tensor_load_to_lds(g0, g1, g2, g3, gz, 0);
#else
  __builtin_amdgcn_tensor_load_to_lds(g0, g1, g2, g3, 0);
#endif
}

// n must fold to a constant (call sites are in a fully unrolled loop).
__device__ __forceinline__ void wait_tensorcnt_n(int n) {
  asm volatile("" ::: "memory");            // compiler fence: no LDS-read motion
  switch (n) {
    case 0:  __builtin_amdgcn_s_wait_tensorcnt(0); break;
    case 1:  __builtin_amdgcn_s_wait_tensorcnt(1); break;
    case 2:  __builtin_amdgcn_s_wait_tensorcnt(2); break;
    default: __builtin_amdgcn_s_wait_tensorcnt(3); break;
  }
  asm volatile("" ::: "memory");
}

__global__ __launch_bounds__(32 * WAVES_PER_BLOCK)
void pep_embed_bag_kernel(const int* __restrict__ x,
                          const float* __restrict__ v,
                          const float* __restrict__ s,
                          float* __restrict__ out) {
  // 4-deep 8-row tiles per wave: 8 waves * 4 * 2KB = 64 KB (<< 320 KB/WGP).
  __shared__ __align__(16) float lbuf[WAVES_PER_BLOCK][NBUF][ROWS_PER_DESC * DIM];

  const int lane = threadIdx.x & 31;
  // Force wave id into the uniform (SGPR) domain so the bag-index loads become
  // scalar SMEM loads and the D# descriptors are built with pure SALU.
  const int wave = __builtin_amdgcn_readfirstlane((int)(threadIdx.x >> 5));
  const int b    = blockIdx.x * WAVES_PER_BLOCK + wave;     // output row

  // Per-lane soft-threshold for columns 2*lane, 2*lane+1 (computed once).
  const float th0 = 1.0f / (1.0f + __expf(-s[2 * lane]));
  const float th1 = 1.0f / (1.0f + __expf(-s[2 * lane + 1]));

  const int* bag = x + (size_t)b * BAG;     // uniform pointer -> s_load path

  // Low 32 bits of a generic LDS pointer == LDS byte offset (aperture math).
  uint32_t ldsb[NBUF];
#pragma unroll
  for (int i = 0; i < NBUF; ++i)
    ldsb[i] = (uint32_t)(uintptr_t)&lbuf[wave][i][0];

  // Prime the pipeline: descriptors 0..2 -> buffers 0..2 (3 in flight).
#pragma unroll
  for (int k0 = 0; k0 < 3; ++k0) {
    int ids[ROWS_PER_DESC];
#pragma unroll
    for (int t = 0; t < ROWS_PER_DESC; ++t) {
      int j = k0 * ROWS_PER_DESC + t;
      ids[t] = bag[j < BAG ? j : BAG - 1];
    }
    tdm_gather8(v, ldsb[k0], ids, (k0 == NDESC - 1) ? LAST_ROWS : ROWS_PER_DESC);
  }

  float accx = 0.0f, accy = 0.0f;

#pragma unroll
  for (int k = 0; k < NDESC; ++k) {
    if (k + 3 < NDESC) {
      const int kn = k + 3;
      const int nv = (kn == NDESC - 1) ? LAST_ROWS : ROWS_PER_DESC;
      int ids[ROWS_PER_DESC];
#pragma unroll
      for (int t = 0; t < ROWS_PER_DESC; ++t) {
        int j = kn * ROWS_PER_DESC + t;
        ids[t] = bag[j < BAG ? j : BAG - 1];   // clamp: avoid OOB read of x
      }
      tdm_gather8(v, ldsb[kn & (NBUF - 1)], ids, nv);
    }
    // In-order TDM completion: descriptor k is done once outstanding <= #issued-after-k.
    {
      const int issued_after_k = ((NDESC - 1) - k) < 3 ? ((NDESC - 1) - k) : 3;
      wait_tensorcnt_n(issued_after_k);        // folds to 3,3,3,3,2,1,0
    }

    const float* buf = &lbuf[wave][k & (NBUF - 1)][0];
    const int nrows = (k == NDESC - 1) ? LAST_ROWS : ROWS_PER_DESC;
#pragma unroll
    for (int r = 0; r < ROWS_PER_DESC; ++r) {
      if (r < nrows) {
        // lane i reads columns 2i,2i+1 -> banks 2i,2i+1: conflict-free ds_load_b64
        float2 val = *reinterpret_cast<const float2*>(&buf[r * DIM + lane * 2]);
        float a0 = fabsf(val.x) - th0;
        float a1 = fabsf(val.y) - th1;
        accx += (a0 > 0.0f) ? copysignf(a0, val.x) : 0.0f;
        accy += (a1 > 0.0f) ? copysignf(a1, val.y) : 0.0f;
      }
    }
  }

  float2 o;
  o.x = accx;
  o.y = accy;
  *reinterpret_cast<float2*>(&out[(size_t)b * DIM + lane * 2]) = o;  // coalesced b64 store
}

extern "C" void kernel_launch(void* const* d_in, const int* in_sizes, int n_in,
                              void* d_out, int out_size, void* d_ws, size_t ws_size,
                              hipStream_t stream) {
  const int*   x = (const int*)d_in[0];    // [4096, 50]
  const float* v = (const float*)d_in[1];  // [1M, 64]
  const float* s = (const float*)d_in[2];  // [64]
  float*     out = (float*)d_out;          // [4096, 64]

  dim3 grid(BATCH / WAVES_PER_BLOCK);      // 512 blocks
  dim3 block(32 * WAVES_PER_BLOCK);        // 256 threads = 8 wave32
  hipLaunchKernelGGL(pep_embed_bag_kernel, grid, block, 0, stream, x, v, s, out);
}